// SAGEBlock_80668075753587
// MI455X (gfx1250) — compile-verified
//
#include <hip/hip_runtime.h>
#include <math.h>

#define DD 128
#define GG 8
#define EPS 1e-5f

typedef __attribute__((ext_vector_type(2))) float v2f;
typedef __attribute__((ext_vector_type(8))) float v8f;

__device__ __forceinline__ float gelu_exact(float v) {
    return 0.5f * v * (1.0f + erff(v * 0.70710678118654752440f));
}

// ---------- 0: zero workspace ----------
__global__ void k_zero(float* p, long n) {
    long i = (long)blockIdx.x * blockDim.x + threadIdx.x;
    long stride = (long)gridDim.x * blockDim.x;
    for (; i < n; i += stride) p[i] = 0.0f;
}

// ---------- 1: edge scatter: agg[dst] += x[src], deg[dst] += 1 ----------
// one thread per (edge, 4-feature chunk): E*32 threads
__global__ void k_edge_scatter(const float* __restrict__ x,
                               const int* __restrict__ src,
                               const int* __restrict__ dst,
                               float* __restrict__ agg,
                               float* __restrict__ deg, int E) {
    long t = (long)blockIdx.x * blockDim.x + threadIdx.x;
    if (t >= (long)E * 32) return;
    int e = (int)(t >> 5);
    int q = (int)(t & 31);
    int s = src[e], d = dst[e];
    const float4 v = *(const float4*)(x + (size_t)s * DD + q * 4);
    float* a = agg + (size_t)d * DD + q * 4;
    atomicAdd(a + 0, v.x);
    atomicAdd(a + 1, v.y);
    atomicAdd(a + 2, v.z);
    atomicAdd(a + 3, v.w);
    if (q == 0) atomicAdd(deg + d, 1.0f);
}

// ---------- 2: per-graph node counts ----------
__global__ void k_count(const int* __restrict__ batch, float* __restrict__ cnt, int N) {
    int i = blockIdx.x * blockDim.x + threadIdx.x;
    if (i < N) atomicAdd(cnt + batch[i], 1.0f);
}

// ---------- 3: fused GEMM (WMMA f32) + bias + GELU + per-graph sum ----------
// block: 256 threads = 8 waves; covers 16 node-rows x all 128 output cols.
// wave w computes the 16x16 tile for columns [16w, 16w+16).
__global__ __launch_bounds__(256) void k_gemm_gelu(
    const float* __restrict__ agg,   // [N,D] (unscaled sums)
    const float* __restrict__ deg,   // [N]
    const float* __restrict__ x,     // [N,D]
    const float* __restrict__ Wl,    // [D,D] row j = output feature
    const float* __restrict__ Wr,    // [D,D]
    const float* __restrict__ bl,    // [D]
    const int* __restrict__ batch,   // [N]
    float* __restrict__ f,           // [N,D] output (pre-norm activations)
    float* __restrict__ gsum,        // [G,D]
    int N) {
    __shared__ float sAgg[16][DD + 4];
    __shared__ float sX[16][DD + 4];
    __shared__ int sBatch[16];

    const int row0 = blockIdx.x * 16;
    const int tid = threadIdx.x;

    // cooperative load: 16 rows * 32 float4 = 512 slots, 2 per thread
    for (int idx = tid; idx < 512; idx += 256) {
        int r = idx >> 5;
        int c4 = (idx & 31) << 2;
        int node = row0 + r;
        float4 av = make_float4(0.f, 0.f, 0.f, 0.f);
        float4 xv = make_float4(0.f, 0.f, 0.f, 0.f);
        if (node < N) {
            float rd = 1.0f / fmaxf(deg[node], 1.0f);
            float4 a = *(const float4*)(agg + (size_t)node * DD + c4);
            av = make_float4(a.x * rd, a.y * rd, a.z * rd, a.w * rd);
            xv = *(const float4*)(x + (size_t)node * DD + c4);
        }
        sAgg[r][c4 + 0] = av.x; sAgg[r][c4 + 1] = av.y;
        sAgg[r][c4 + 2] = av.z; sAgg[r][c4 + 3] = av.w;
        sX[r][c4 + 0] = xv.x; sX[r][c4 + 1] = xv.y;
        sX[r][c4 + 2] = xv.z; sX[r][c4 + 3] = xv.w;
    }
    if (tid < 16) {
        int node = row0 + tid;
        sBatch[tid] = (node < N) ? batch[node] : 0;
    }
    __syncthreads();

    const int w = tid >> 5;          // wave id: column tile
    const int lane = tid & 31;
    const int m = lane & 15;         // A row / C column index
    const int hi = lane >> 4;        // k-pair selector / C row-half
    const int j = (w << 4) + m;      // global output feature (B column n, C column n)

    const float* WlRow = Wl + (size_t)j * DD;  // B[k, j] = Wl[j, k]
    const float* WrRow = Wr + (size_t)j * DD;

    v8f c = {};
    #pragma unroll
    for (int kb = 0; kb < DD; kb += 4) {
        const int kA = kb + (hi << 1);
        v2f a;  a.x = sAgg[m][kA]; a.y = sAgg[m][kA + 1];
        v2f b;  b.x = WlRow[kA];   b.y = WlRow[kA + 1];
        c = __builtin_amdgcn_wmma_f32_16x16x4_f32(false, a, false, b,
                                                  (short)0, c, false, false);
        v2f a2; a2.x = sX[m][kA];  a2.y = sX[m][kA + 1];
        v2f b2; b2.x = WrRow[kA];  b2.y = WrRow[kA + 1];
        c = __builtin_amdgcn_wmma_f32_16x16x4_f32(false, a2, false, b2,
                                                  (short)0, c, false, false);
    }

    const float blj = bl[j];
    #pragma unroll
    for (int v = 0; v < 8; ++v) {
        int mrow = v + (hi << 3);        // C/D row for this VGPR
        int node = row0 + mrow;
        if (node < N) {
            float val = gelu_exact(c[v] + blj);
            f[(size_t)node * DD + j] = val;
            atomicAdd(gsum + (size_t)sBatch[mrow] * DD + j, val);
        }
    }
}

// ---------- 4: mean = gsum / cnt (in place) ----------
__global__ void k_mean(float* __restrict__ gsum, const float* __restrict__ cnt) {
    int i = blockIdx.x * blockDim.x + threadIdx.x;
    if (i < GG * DD) gsum[i] /= cnt[i / DD];
}

// ---------- 5: center + accumulate sum of squares ----------
// one thread per (node, 4-feature chunk)
__global__ void k_center(float* __restrict__ f,               // in: f, out: out0
                         const float* __restrict__ mean,      // [G,D]
                         const float* __restrict__ mean_scale,// [D]
                         const int* __restrict__ batch,
                         float* __restrict__ gvar, int N) {
    long t = (long)blockIdx.x * blockDim.x + threadIdx.x;
    if (t >= (long)N * 32) return;
    int i = (int)(t >> 5);
    int q4 = (int)(t & 31) << 2;
    int g = batch[i];
    float4 fv = *(float4*)(f + (size_t)i * DD + q4);
    const float4 mv = *(const float4*)(mean + (size_t)g * DD + q4);
    const float4 ms = *(const float4*)(mean_scale + q4);
    float4 o;
    o.x = fv.x - ms.x * mv.x;
    o.y = fv.y - ms.y * mv.y;
    o.z = fv.z - ms.z * mv.z;
    o.w = fv.w - ms.w * mv.w;
    *(float4*)(f + (size_t)i * DD + q4) = o;
    float* gv = gvar + (size_t)g * DD + q4;
    atomicAdd(gv + 0, o.x * o.x);
    atomicAdd(gv + 1, o.y * o.y);
    atomicAdd(gv + 2, o.z * o.z);
    atomicAdd(gv + 3, o.w * o.w);
}

// ---------- 6: inv = rsqrt(var/cnt + eps) (in place in gvar) ----------
__global__ void k_var(float* __restrict__ gvar, const float* __restrict__ cnt) {
    int i = blockIdx.x * blockDim.x + threadIdx.x;
    if (i < GG * DD) gvar[i] = rsqrtf(gvar[i] / cnt[i / DD] + EPS);
}

// ---------- 7: normalize + affine + residual ----------
__global__ void k_final(float* __restrict__ out0,             // in: out0, out: result
                        const float* __restrict__ inv,        // [G,D]
                        const float* __restrict__ gn_w,
                        const float* __restrict__ gn_b,
                        const float* __restrict__ x,
                        const int* __restrict__ batch, int N) {
    long t = (long)blockIdx.x * blockDim.x + threadIdx.x;
    if (t >= (long)N * 32) return;
    int i = (int)(t >> 5);
    int q4 = (int)(t & 31) << 2;
    int g = batch[i];
    float4 o = *(float4*)(out0 + (size_t)i * DD + q4);
    const float4 iv = *(const float4*)(inv + (size_t)g * DD + q4);
    const float4 wv = *(const float4*)(gn_w + q4);
    const float4 bv = *(const float4*)(gn_b + q4);
    const float4 xv = *(const float4*)(x + (size_t)i * DD + q4);
    o.x = o.x * iv.x * wv.x + bv.x + xv.x;
    o.y = o.y * iv.y * wv.y + bv.y + xv.y;
    o.z = o.z * iv.z * wv.z + bv.z + xv.z;
    o.w = o.w * iv.w * wv.w + bv.w + xv.w;
    *(float4*)(out0 + (size_t)i * DD + q4) = o;
}

extern "C" void kernel_launch(void* const* d_in, const int* in_sizes, int n_in,
                              void* d_out, int out_size, void* d_ws, size_t ws_size,
                              hipStream_t stream) {
    const float* x  = (const float*)d_in[0];
    const int* ei   = (const int*)d_in[1];     // [2,E]: src then dst
    const int* batch= (const int*)d_in[2];
    const float* Wl = (const float*)d_in[4];
    const float* bl = (const float*)d_in[5];
    const float* Wr = (const float*)d_in[6];
    const float* gn_w = (const float*)d_in[7];
    const float* gn_b = (const float*)d_in[8];
    const float* mscale = (const float*)d_in[9];

    const int N = in_sizes[0] / DD;
    const int E = in_sizes[1] / 2;
    const int* src = ei;
    const int* dst = ei + E;

    float* f = (float*)d_out;                 // reused: f -> out0 -> result

    // workspace layout (floats)
    float* agg  = (float*)d_ws;               // N*D
    float* deg  = agg + (size_t)N * DD;       // N
    float* gsum = deg + N;                    // G*D
    float* gvar = gsum + GG * DD;             // G*D
    float* cnt  = gvar + GG * DD;             // G
    const long zeroN = (long)N * DD + N + 2L * GG * DD + GG;

    k_zero<<<2048, 256, 0, stream>>>((float*)d_ws, zeroN);

    {
        long tot = (long)E * 32;
        int blocks = (int)((tot + 255) / 256);
        k_edge_scatter<<<blocks, 256, 0, stream>>>(x, src, dst, agg, deg, E);
    }

    k_count<<<(N + 255) / 256, 256, 0, stream>>>(batch, cnt, N);

    {
        int rowBlocks = (N + 15) / 16;
        k_gemm_gelu<<<rowBlocks, 256, 0, stream>>>(agg, deg, x, Wl, Wr, bl,
                                                   batch, f, gsum, N);
    }

    k_mean<<<(GG * DD + 255) / 256, 256, 0, stream>>>(gsum, cnt);

    {
        long tot = (long)N * 32;
        int blocks = (int)((tot + 255) / 256);
        k_center<<<blocks, 256, 0, stream>>>(f, gsum, mscale, batch, gvar, N);
    }

    k_var<<<(GG * DD + 255) / 256, 256, 0, stream>>>(gvar, cnt);

    {
        long tot = (long)N * 32;
        int blocks = (int)((tot + 255) / 256);
        k_final<<<blocks, 256, 0, stream>>>(f, gvar, gn_w, gn_b, x, batch, N);
    }
}